// QuantSoftmax_16269336117398
// MI455X (gfx1250) — compile-verified
//
#include <hip/hip_runtime.h>
#include <math.h>

// Problem shape (fixed by the reference): B=1, H=12, S=2048
#define SLEN   2048
#define NROWS  (12 * 2048)
#define NELEM  ((size_t)NROWS * (size_t)SLEN)

typedef __attribute__((ext_vector_type(2))) float v2f;
typedef __attribute__((ext_vector_type(8))) float v8f;

// ---- ordered-uint encoding so float min/max can use integer atomics ----
__device__ __forceinline__ unsigned f32_ord(float f) {
  unsigned u = __float_as_uint(f);
  return (u & 0x80000000u) ? ~u : (u | 0x80000000u);
}
__device__ __forceinline__ float ord_f32(unsigned u) {
  return __uint_as_float((u & 0x80000000u) ? (u ^ 0x80000000u) : ~u);
}

// symmetric_linear_quantization_params (16-bit): clamp(max(|mn|,|mx|),1e-8)/32767
__device__ __forceinline__ float sym_scale16(float mn, float mx) {
  return fmaxf(fmaxf(fabsf(mn), fabsf(mx)), 1e-8f) / 32767.0f;
}

// ---- CDNA5 async data path: stage one 8KB row (2048 f32) into LDS ----
// Each of the 256 threads copies 32 bytes with two GLOBAL_LOAD_ASYNC_TO_LDS_B128
// (INST_OFFSET is added to both the LDS and global addresses per the ISA), then
// waits on ASYNCcnt and barriers so all waves see the staged row.
__device__ __forceinline__ void async_row_to_lds(const float* __restrict__ gsrc,
                                                 float* lds_buf) {
  unsigned loff = (unsigned)(size_t)(void*)lds_buf + (unsigned)(threadIdx.x * 32u);
  unsigned long long g =
      (unsigned long long)(size_t)gsrc + (unsigned long long)(threadIdx.x * 32u);
  asm volatile("global_load_async_to_lds_b128 %0, %1, off\n\t"
               "global_load_async_to_lds_b128 %0, %1, off offset:16"
               :: "v"(loff), "v"(g) : "memory");
  asm volatile("s_wait_asynccnt 0x0" ::: "memory");
  __syncthreads();
}

// ---------------------------------------------------------------- kernel 0
__global__ void k_init(unsigned* uws) {
  if (threadIdx.x == 0 && blockIdx.x == 0) {
    uws[0] = 0xFFFFFFFFu;  // min(x)   (ordered encoding)
    uws[1] = 0u;           // max(x)
    uws[2] = 0xFFFFFFFFu;  // min(exp_int)
    uws[3] = 0u;           // max(exp_int)
  }
}

// ---------------------------------------------------------------- kernel 1
// Global min/max of x (f32), float4 grid-stride + LDS tree + integer atomics.
__global__ void __launch_bounds__(256) k_minmax(const float* __restrict__ x,
                                                unsigned* __restrict__ uws,
                                                size_t n4) {
  __shared__ float r0[256];
  __shared__ float r1[256];
  const int tid = threadIdx.x;
  size_t i = (size_t)blockIdx.x * blockDim.x + tid;
  const size_t stride = (size_t)gridDim.x * blockDim.x;
  float mn = 3.402823466e38f, mx = -3.402823466e38f;
  const float4* x4 = (const float4*)x;
  for (; i < n4; i += stride) {
    float4 v = x4[i];
    mn = fminf(mn, fminf(fminf(v.x, v.y), fminf(v.z, v.w)));
    mx = fmaxf(mx, fmaxf(fmaxf(v.x, v.y), fmaxf(v.z, v.w)));
  }
  r0[tid] = mn; r1[tid] = mx;
  __syncthreads();
  for (int s = 128; s > 0; s >>= 1) {
    if (tid < s) {
      r0[tid] = fminf(r0[tid], r0[tid + s]);
      r1[tid] = fmaxf(r1[tid], r1[tid + s]);
    }
    __syncthreads();
  }
  if (tid == 0) {
    atomicMin(&uws[0], f32_ord(r0[0]));
    atomicMax(&uws[1], f32_ord(r1[0]));
  }
}

// ---------------------------------------------------------------- kernel 2
// Per row: quantize, row max (stored to ws), int_exp, global exp min/max.
__global__ void __launch_bounds__(256) k_exp_pass(const float* __restrict__ x,
                                                  unsigned* __restrict__ uws,
                                                  float* __restrict__ rowmax) {
  __shared__ float buf[SLEN];
  __shared__ float r0[256];
  __shared__ float r1[256];
  const int tid = threadIdx.x;
  const int row = blockIdx.x;
  const float* src = x + (size_t)row * SLEN;

  const float sf  = sym_scale16(ord_f32(uws[0]), ord_f32(uws[1]));
  const float x0i = floorf(-0.6931f / sf);                              // floor(-ln2/sf) < 0
  const float bi  = floorf((0.96963238f / 0.35815147f) / sf);
  const float ci  = floorf((1.0f / 0.35815147f) / (sf * sf));

  async_row_to_lds(src, buf);

  // quantize (mimic reference: xi = (round(clip(x/sf)) * sf) / sf ) + row max
  float xi[8];
  float m = -3.402823466e38f;
#pragma unroll
  for (int k = 0; k < 8; ++k) {
    float v = buf[tid + 256 * k];
    float q = nearbyintf(v / sf);                    // jnp.round = RNE
    q = fminf(fmaxf(q, -32768.0f), 32767.0f);
    float xq = (q * sf) / sf;                        // QuantAct dequant/requant round-trip
    xi[k] = xq;
    m = fmaxf(m, xq);
  }
  r0[tid] = m;
  __syncthreads();
  for (int s = 128; s > 0; s >>= 1) {
    if (tid < s) r0[tid] = fmaxf(r0[tid], r0[tid + s]);
    __syncthreads();
  }
  const float rmax = r0[0];
  __syncthreads();                                    // r0 is reused below
  if (tid == 0) rowmax[row] = rmax;

  // int_exp + local min/max of exp_int
  float lmin = 3.402823466e38f, lmax = -3.402823466e38f;
#pragma unroll
  for (int k = 0; k < 8; ++k) {
    float t = xi[k] - rmax;
    t = fmaxf(t, 30.0f * x0i);
    float q = floorf(t / x0i);                        // q in [0, 30]
    float r = t - x0i * q;                            // r in (x0i, 0]
    float z = r * (r + bi) + ci;
    float e = fmaxf(floorf(ldexpf(z, 30 - (int)q)), 0.0f);
    lmin = fminf(lmin, e);
    lmax = fmaxf(lmax, e);
  }
  r0[tid] = lmin; r1[tid] = lmax;
  __syncthreads();
  for (int s = 128; s > 0; s >>= 1) {
    if (tid < s) {
      r0[tid] = fminf(r0[tid], r0[tid + s]);
      r1[tid] = fmaxf(r1[tid], r1[tid + s]);
    }
    __syncthreads();
  }
  if (tid == 0) {
    atomicMin(&uws[2], f32_ord(r0[0]));
    atomicMax(&uws[3], f32_ord(r1[0]));
  }
}

// ---------------------------------------------------------------- kernel 3
// Per row: recompute exp_int, fixedpoint_mul requantize (f64, frexp/31-bit
// mantissa exactly as I-BERT), row sum via V_WMMA_F32_16X16X4_F32, emit output.
__global__ void __launch_bounds__(256) k_out_pass(const float* __restrict__ x,
                                                  const unsigned* __restrict__ uws,
                                                  const float* __restrict__ rowmax,
                                                  float* __restrict__ out) {
  __shared__ float buf[SLEN];
  __shared__ float red[256];
  const int tid = threadIdx.x;
  const int row = blockIdx.x;
  const float* src = x + (size_t)row * SLEN;

  const float sf  = sym_scale16(ord_f32(uws[0]), ord_f32(uws[1]));
  const float x0i = floorf(-0.6931f / sf);
  const float bi  = floorf((0.96963238f / 0.35815147f) / sf);
  const float ci  = floorf((1.0f / 0.35815147f) / (sf * sf));
  const float exp_sf = (0.35815147f * sf * sf) / 1073741824.0f;         // / 2^30
  const float act_sf = sym_scale16(ord_f32(uws[2]), ord_f32(uws[3]));

  // fixedpoint_mul constants: new_scale = exp_sf/act_sf (f32), then f64 frexp:
  // e = floor(log2|ns|)+1 ; m_int = round(ns/2^e * 2^31) ; out = round(z_int*m_int/2^(31-e))
  const double ns = (double)(exp_sf / act_sf);
  int e2;
  const double mant   = frexp(ns, &e2);
  const double m_int  = nearbyint(ldexp(mant, 31));
  const double qscale = ldexp(m_int, e2 - 31);   // exact: 31-bit int * power of 2
  const double dsf    = (double)exp_sf;

  async_row_to_lds(src, buf);
  const float rmax = rowmax[row];

  int e16[8];
  int psum = 0;
#pragma unroll
  for (int k = 0; k < 8; ++k) {
    float v = buf[tid + 256 * k];
    float q0 = nearbyintf(v / sf);
    q0 = fminf(fmaxf(q0, -32768.0f), 32767.0f);
    float xq = (q0 * sf) / sf;
    float t = xq - rmax;
    t = fmaxf(t, 30.0f * x0i);
    float q = floorf(t / x0i);
    float r = t - x0i * q;
    float z = r * (r + bi) + ci;
    float ei = fmaxf(floorf(ldexpf(z, 30 - (int)q)), 0.0f);
    double z_int = nearbyint((double)ei / dsf);
    double o = nearbyint(z_int * qscale);
    o = fmin(fmax(o, -32768.0), 32767.0);
    e16[k] = (int)o;
    psum += e16[k];
  }

  // ---- row sum: 256 exact f32 partials reduced on the matrix unit ----
  red[tid] = (float)psum;   // psum <= 8*32767 < 2^24 -> exact in f32
  __syncthreads();
#if __has_builtin(__builtin_amdgcn_wmma_f32_16x16x4_f32)
  if (tid < 32) {           // wave 0, EXEC all ones (uniform branch)
    v2f a; a[0] = 1.0f; a[1] = 1.0f;           // A = ones(16x4)
    v8f acc = {};
#pragma unroll
    for (int j = 0; j < 4; ++j) {
      // B(4x16) layout: VGPR0 lanes0-15=B[0][n], lanes16-31=B[1][n]; VGPR1=B[2],B[3]
      v2f b; b[0] = red[tid + 64 * j]; b[1] = red[tid + 32 + 64 * j];
      acc = __builtin_amdgcn_wmma_f32_16x16x4_f32(
          false, a, false, b, (short)0, acc, false, false);
    }
    red[tid] = acc[0];      // lanes 0..15: column sums (partials mod 16)
  }
  __syncthreads();
  if (tid == 0) {
    float t = 0.0f;
    for (int n = 0; n < 16; ++n) t += red[n];
    red[0] = t;
  }
  __syncthreads();
  const float ssum = red[0];
#else
  for (int s = 128; s > 0; s >>= 1) {
    if (tid < s) red[tid] += red[tid + s];
    __syncthreads();
  }
  const float ssum = red[0];
#endif

  // factor = floor(2^32 / s) in f32 (weak-typed promotion in the reference)
  const float factor = floorf(4294967296.0f / ssum);
  float* dst = out + (size_t)row * SLEN;
#pragma unroll
  for (int k = 0; k < 8; ++k) {
    float oi = floorf(((float)e16[k] * factor) * (1.0f / 16777216.0f)); // / 2^24
    dst[tid + 256 * k] = oi * (1.0f / 256.0f);                          // * out_sf
  }
}

// ---------------------------------------------------------------- launch
extern "C" void kernel_launch(void* const* d_in, const int* in_sizes, int n_in,
                              void* d_out, int out_size, void* d_ws, size_t ws_size,
                              hipStream_t stream) {
  const float* x = (const float*)d_in[0];
  float* out = (float*)d_out;
  unsigned* uws = (unsigned*)d_ws;          // uws[0..3]: atomic min/max slots
  float* rowmax = (float*)d_ws + 16;        // 24576 per-row maxima

  k_init<<<1, 32, 0, stream>>>(uws);
  k_minmax<<<2048, 256, 0, stream>>>(x, uws, NELEM / 4);
  k_exp_pass<<<NROWS, 256, 0, stream>>>(x, uws, rowmax);
  k_out_pass<<<NROWS, 256, 0, stream>>>(x, uws, rowmax, out);
}